// LSTM_37048387895664
// MI455X (gfx1250) — compile-verified
//
#include <hip/hip_runtime.h>

#define S_LEN 2048
#define B_SZ  4096
#define D_IN  12
#define TPW   2          // batch tiles (16 rows each) interleaved per wave

typedef __attribute__((ext_vector_type(2))) float v2f;
typedef __attribute__((ext_vector_type(8))) float v8f;

// CDNA5 has native V_TANH_F32 (TRANS pipe) -> 1-instr tanh, 3-instr sigmoid.
#if __has_builtin(__builtin_amdgcn_tanhf)
__device__ __forceinline__ float tanh_f(float x) { return __builtin_amdgcn_tanhf(x); }
#else
__device__ __forceinline__ float tanh_f(float x) {
    return 2.0f * __builtin_amdgcn_rcpf(1.0f + __expf(-2.0f * x)) - 1.0f;
}
#endif
__device__ __forceinline__ float sigm(float x) { return fmaf(tanh_f(0.5f * x), 0.5f, 0.5f); }

// Transposed formulation: gates^T = W_perm x X^T  (M = permuted gate, N = batch row, K = features).
// Gate-row permutation perm(p) = (p&3)*4 + (p>>2) puts, for every lane, the i/f/g/o of its two
// hidden components into acc[0..7] directly (lanes<16: j=0,1; lanes>=16: j=2,3), and the pointwise
// output (h_a,h_b) is already the B fragment of h^T for the next K=4 recurrent WMMA.
__global__ __launch_bounds__(256) void lstm2_fused(
    const float* __restrict__ x,
    const float* __restrict__ Wih0, const float* __restrict__ Whh0,
    const float* __restrict__ bih0, const float* __restrict__ bhh0,
    const float* __restrict__ Wih1, const float* __restrict__ Whh1,
    const float* __restrict__ bih1, const float* __restrict__ bhh1,
    const float* __restrict__ Wfc,  const float* __restrict__ bfc,
    float* __restrict__ out)
{
    const int lane = threadIdx.x & 31;
    const int wave = threadIdx.x >> 5;
    const int half = lane >> 4;            // K-half for A/B fragments
    const int row  = lane & 15;            // batch row within tile (as B-matrix N)
    const int k2   = 2 * half;             // first K index held by this lane
    const int p    = lane & 15;            // permuted gate row (as A-matrix M)
    const int n    = ((p & 3) << 2) | (p >> 2);   // original torch gate row

    // ---- loop-invariant A fragments (weights, permuted rows) ----
    v2f aX[3];
#pragma unroll
    for (int c = 0; c < 3; ++c) {
        aX[c].x = Wih0[n * D_IN + 4 * c + k2];
        aX[c].y = Wih0[n * D_IN + 4 * c + k2 + 1];
    }
    v2f aH0, aI1, aH1;
    aH0.x = Whh0[n * 4 + k2]; aH0.y = Whh0[n * 4 + k2 + 1];
    aI1.x = Wih1[n * 4 + k2]; aI1.y = Wih1[n * 4 + k2 + 1];
    aH1.x = Whh1[n * 4 + k2]; aH1.y = Whh1[n * 4 + k2 + 1];

    // ---- bias C-operands (per-lane: gate perm(r + 8*half)) ----
    v8f bC0, bC1;
#pragma unroll
    for (int r = 0; r < 8; ++r) {
        const int pr = r + 8 * half;
        const int nr = ((pr & 3) << 2) | (pr >> 2);
        bC0[r] = bih0[nr] + bhh0[nr];
        bC1[r] = bih1[nr] + bhh1[nr];
    }
    const float wfA = Wfc[k2], wfB = Wfc[k2 + 1];
    const float bias_fc = bfc[0];

    // ---- two interleaved batch tiles per wave; 32-bit element offsets keep
    //      loads on the GLOBAL (saddr + 32-bit voffset) path, not FLAT ----
    const int tile0 = (blockIdx.x * 8 + wave) * TPW;
    const unsigned xstep = (unsigned)B_SZ * D_IN;   // elements per timestep
    unsigned xoff[TPW];                             // lane's fragment base at time t
    unsigned orow[TPW];                             // lane's output column
#pragma unroll
    for (int u = 0; u < TPW; ++u) {
        const unsigned baseRow = (unsigned)(tile0 + u) * 16u;
        xoff[u] = (baseRow + (unsigned)row) * D_IN + (unsigned)k2;
        orow[u] = baseRow + (unsigned)row;
    }

    // recurrent state (per tile, per lane: its two hidden components)
    float c0a[TPW], c0b[TPW], c1a[TPW], c1b[TPW];
    v2f h1[TPW], h2[TPW];
#pragma unroll
    for (int u = 0; u < TPW; ++u) {
        c0a[u] = c0b[u] = c1a[u] = c1b[u] = 0.f;
        h1[u].x = h1[u].y = 0.f; h2[u].x = h2[u].y = 0.f;
    }

    // B fragments of x^T for t=0 (chunk c: K = 4c + k2 + {0,1})
    v2f bx[TPW][3];
#pragma unroll
    for (int u = 0; u < TPW; ++u) {
        bx[u][0] = *(const v2f*)(x + xoff[u] + 0);
        bx[u][1] = *(const v2f*)(x + xoff[u] + 4);
        bx[u][2] = *(const v2f*)(x + xoff[u] + 8);
    }

    unsigned ooff = 0;                              // t * B_SZ, incremented
    for (int t = 0; t < S_LEN; ++t) {
        // uniform step deltas (clamped at sequence end) -> no per-step multiplies
        const unsigned d1 = (t + 1 < S_LEN) ? xstep : 0u;
        const unsigned d2 = (t + 2 < S_LEN) ? xstep : 0u;

        v2f nbx[TPW][3];
#pragma unroll
        for (int u = 0; u < TPW; ++u) {
            const unsigned pn = xoff[u] + d1;       // next timestep's fragments
            nbx[u][0] = *(const v2f*)(x + pn + 0);
            nbx[u][1] = *(const v2f*)(x + pn + 4);
            nbx[u][2] = *(const v2f*)(x + pn + 8);
            __builtin_prefetch(x + pn + d2, 0, 1);  // pull t+2 toward L2
            xoff[u] = pn;
        }

#pragma unroll
        for (int u = 0; u < TPW; ++u) {
            // ---- layer 0: gates0^T = Wih0_p * x^T (+bias) + Whh0_p * h1^T  (4 WMMAs) ----
            v8f g0;
            g0 = __builtin_amdgcn_wmma_f32_16x16x4_f32(false, aX[0], false, bx[u][0], (short)0, bC0, false, false);
            g0 = __builtin_amdgcn_wmma_f32_16x16x4_f32(false, aX[1], false, bx[u][1], (short)0, g0,  false, false);
            g0 = __builtin_amdgcn_wmma_f32_16x16x4_f32(false, aX[2], false, bx[u][2], (short)0, g0,  false, false);
            g0 = __builtin_amdgcn_wmma_f32_16x16x4_f32(false, aH0,   false, h1[u],    (short)0, g0,  false, false);

            // pointwise: g0[0..3] = i,f,g,o of component j=2*half; g0[4..7] = j=2*half+1
            float ca = sigm(g0[1]) * c0a[u] + sigm(g0[0]) * tanh_f(g0[2]);
            float ha = sigm(g0[3]) * tanh_f(ca);
            float cb = sigm(g0[5]) * c0b[u] + sigm(g0[4]) * tanh_f(g0[6]);
            float hb = sigm(g0[7]) * tanh_f(cb);
            c0a[u] = ca; c0b[u] = cb;
            v2f h1n; h1n.x = ha; h1n.y = hb;        // == B fragment of h1^T

            // ---- layer 1: gates1^T = Wih1_p * h1^T (+bias) + Whh1_p * h2^T  (2 WMMAs) ----
            v8f g1;
            g1 = __builtin_amdgcn_wmma_f32_16x16x4_f32(false, aI1, false, h1n,   (short)0, bC1, false, false);
            g1 = __builtin_amdgcn_wmma_f32_16x16x4_f32(false, aH1, false, h2[u], (short)0, g1,  false, false);

            float da  = sigm(g1[1]) * c1a[u] + sigm(g1[0]) * tanh_f(g1[2]);
            float h2a = sigm(g1[3]) * tanh_f(da);
            float db  = sigm(g1[5]) * c1b[u] + sigm(g1[4]) * tanh_f(g1[6]);
            float h2b = sigm(g1[7]) * tanh_f(db);
            c1a[u] = da; c1b[u] = db;
            h1[u] = h1n;
            h2[u].x = h2a; h2[u].y = h2b;

            // ---- fc: out[row] = b_fc + sum_q h2[row][q]*Wfc[q] (cross-half reduce) ----
            float part = h2a * wfA + h2b * wfB;
            float tot  = part + __shfl_xor(part, 16, 32) + bias_fc;
            if (lane < 16)
                out[ooff + orow[u]] = tot;

            bx[u][0] = nbx[u][0]; bx[u][1] = nbx[u][1]; bx[u][2] = nbx[u][2];
        }
        ooff += (unsigned)B_SZ;
    }
}

extern "C" void kernel_launch(void* const* d_in, const int* in_sizes, int n_in,
                              void* d_out, int out_size, void* d_ws, size_t ws_size,
                              hipStream_t stream) {
    (void)in_sizes; (void)n_in; (void)out_size; (void)d_ws; (void)ws_size;
    const float* x    = (const float*)d_in[0];
    const float* Wih0 = (const float*)d_in[1];
    const float* Whh0 = (const float*)d_in[2];
    const float* bih0 = (const float*)d_in[3];
    const float* bhh0 = (const float*)d_in[4];
    const float* Wih1 = (const float*)d_in[5];
    const float* Whh1 = (const float*)d_in[6];
    const float* bih1 = (const float*)d_in[7];
    const float* bhh1 = (const float*)d_in[8];
    const float* Wfc  = (const float*)d_in[9];
    const float* bfc  = (const float*)d_in[10];
    float* out = (float*)d_out;

    // 256 tiles of 16 rows; TPW tiles per wave, 8 waves per block
    dim3 grid(B_SZ / (128 * TPW));   // 16 blocks
    dim3 block(256);
    lstm2_fused<<<grid, block, 0, stream>>>(x, Wih0, Whh0, bih0, bhh0,
                                            Wih1, Whh1, bih1, bhh1,
                                            Wfc, bfc, out);
}